// tGCN_88072599372111
// MI455X (gfx1250) — compile-verified
//
#include <hip/hip_runtime.h>
#include <hip/hip_bf16.h>

#define N_NODES 8192
#define F_IN_DIM 128
#define HID 32
#define NCLS 10

typedef __attribute__((ext_vector_type(2))) float v2f;
typedef __attribute__((ext_vector_type(8))) float v8f;

// ---------------------------------------------------------------- utilities
__global__ void k_zero(float* __restrict__ p, int n) {
  int i = blockIdx.x * blockDim.x + threadIdx.x;
  if (i < n) p[i] = 0.0f;
}

__global__ void k_deg(float* __restrict__ deg, const int* __restrict__ dst, int E) {
  int e = blockIdx.x * blockDim.x + threadIdx.x;
  if (e < E) atomicAdd(&deg[dst[e]], 1.0f);
}

__global__ void k_dinv(float* __restrict__ deg, int n) {
  int i = blockIdx.x * blockDim.x + threadIdx.x;
  if (i < n) deg[i] = rsqrtf(deg[i] + 1.0f);  // +1 = self loop
}

__global__ void k_bias_fill(float* __restrict__ C, const float* __restrict__ bias,
                            int M, int Nc) {
  int t = blockIdx.x * blockDim.x + threadIdx.x;
  if (t < M * Nc) C[t] = bias[t % Nc];
}

__global__ void k_add(float* __restrict__ o, const float* __restrict__ a,
                      const float* __restrict__ b, int n) {
  int i = blockIdx.x * blockDim.x + threadIdx.x;
  if (i < n) o[i] = a[i] + b[i];
}

// ------------------------------------------------- generic f32 WMMA GEMM
// C[M,Nc] = A[M,K] @ B[K,Nc] (+bias), row-major, dims multiples of 16.
// One wave (blockDim=32) computes a 16 x (16*NT) output tile, reusing the A
// registers across NT column tiles. KC = compile-time K-chunk so the loop is
// software-pipelined. grid = (M/16, Nc/(16*NT), K/KC).
// ATOMIC: accumulate into pre-filled C (split-K); else direct store + bias.
template <int KC, int NT, bool ATOMIC>
__global__ void k_gemm(float* __restrict__ C, const float* __restrict__ A,
                       const float* __restrict__ B, const float* __restrict__ bias,
                       int Nc, int K) {
  const int lane = threadIdx.x;            // 0..31, EXEC all-1s (no divergence)
  const int l15  = lane & 15;
  const int hi   = lane >> 4;              // 0 or 1
  const int row  = blockIdx.x * 16 + l15;  // A row held by this lane
  const int col  = blockIdx.y * (16 * NT) + l15;  // first B/C column
  const int koff = hi * 2;                 // lanes 16..31 hold K=2,3 of each step
  const int k0   = blockIdx.z * KC;

  const float* __restrict__ Ar = A + (size_t)row * K + k0 + koff;
  const float* __restrict__ Bc = B + (size_t)(k0 + koff) * Nc + col;

  v8f acc0 = {};
  v8f acc1 = {};
#pragma unroll 4
  for (int kk = 0; kk < KC; kk += 4) {
    if ((kk & 31) == 0)  // pull the streaming A rows ahead into cache
      __builtin_prefetch(Ar + kk + 128, 0, 1);
    v2f a;
    a.x = Ar[kk];
    a.y = Ar[kk + 1];
    v2f b0;
    b0.x = Bc[(size_t)kk * Nc];
    b0.y = Bc[(size_t)(kk + 1) * Nc];
    acc0 = __builtin_amdgcn_wmma_f32_16x16x4_f32(false, a, false, b0, (short)0,
                                                 acc0, false, false);
    if (NT == 2) {
      v2f b1;
      b1.x = Bc[(size_t)kk * Nc + 16];
      b1.y = Bc[(size_t)(kk + 1) * Nc + 16];
      acc1 = __builtin_amdgcn_wmma_f32_16x16x4_f32(false, a, false, b1, (short)0,
                                                   acc1, false, false);
    }
  }

  // C/D layout: VGPR v of lane -> C[blockIdx.x*16 + v + 8*hi][col]
  const int rbase = blockIdx.x * 16 + hi * 8;
  if (ATOMIC) {
#pragma unroll
    for (int v = 0; v < 8; ++v) {
      atomicAdd(&C[(size_t)(rbase + v) * Nc + col], acc0[v]);
      if (NT == 2) atomicAdd(&C[(size_t)(rbase + v) * Nc + col + 16], acc1[v]);
    }
  } else {
    const float bd0 = bias ? bias[col] : 0.0f;
    const float bd1 = (NT == 2 && bias) ? bias[col + 16] : 0.0f;
#pragma unroll
    for (int v = 0; v < 8; ++v) {
      C[(size_t)(rbase + v) * Nc + col] = acc0[v] + bd0;
      if (NT == 2) C[(size_t)(rbase + v) * Nc + col + 16] = acc1[v] + bd1;
    }
  }
}

// ----------------------------------------------------------- GCN aggregation
// out[i,:] = dinv[i]^2 * h[i,:] + bias  (self-loop term + conv bias)
__global__ void k_self(float* __restrict__ out, const float* __restrict__ h,
                       const float* __restrict__ dinv, const float* __restrict__ bias,
                       int N, int H) {
  int t = blockIdx.x * blockDim.x + threadIdx.x;
  if (t >= N * H) return;
  int i = t / H, f = t % H;
  float di = dinv[i];
  out[t] = di * di * h[t] + bias[f];
}

// one wave per edge, lane = feature: out[dst,:] += dinv[s]*dinv[d]*h[src,:]
__global__ void k_edge(float* __restrict__ out, const float* __restrict__ h,
                       const float* __restrict__ dinv, const int* __restrict__ src,
                       const int* __restrict__ dst, int E, int H) {
  size_t t = (size_t)blockIdx.x * blockDim.x + threadIdx.x;
  if (t >= (size_t)E * H) return;
  int e = (int)(t >> 5);  // H == 32
  int f = (int)(t & 31);
  int s = src[e], d = dst[e];
  float nrm = dinv[s] * dinv[d];
  atomicAdd(&out[(size_t)d * H + f], nrm * h[(size_t)s * H + f]);
}

// --------------------------------------------------- BatchNorm(train) + ReLU
// one block per feature column; biased variance over N rows
__global__ void k_bn_relu(float* __restrict__ out, const float* __restrict__ in,
                          const float* __restrict__ g, const float* __restrict__ be,
                          int N, int H) {
  const int f = blockIdx.x;
  __shared__ float ssum[256];
  __shared__ float ssq[256];
  float s = 0.0f, q = 0.0f;
  for (int i = threadIdx.x; i < N; i += blockDim.x) {
    float v = in[(size_t)i * H + f];
    s += v;
    q += v * v;
  }
  ssum[threadIdx.x] = s;
  ssq[threadIdx.x] = q;
  __syncthreads();
  for (int off = blockDim.x >> 1; off > 0; off >>= 1) {
    if (threadIdx.x < off) {
      ssum[threadIdx.x] += ssum[threadIdx.x + off];
      ssq[threadIdx.x]  += ssq[threadIdx.x + off];
    }
    __syncthreads();
  }
  float m   = ssum[0] / (float)N;
  float var = ssq[0] / (float)N - m * m;
  float scale = g[f] * rsqrtf(var + 1e-5f);
  float shift = be[f] - m * scale;
  for (int i = threadIdx.x; i < N; i += blockDim.x) {
    float v = in[(size_t)i * H + f] * scale + shift;
    out[(size_t)i * H + f] = v > 0.0f ? v : 0.0f;
  }
}

// ---------------------------------------- jump classifier + log_softmax (C=10)
__global__ void k_cls(float* __restrict__ out, const float* __restrict__ x0,
                      const float* __restrict__ x1, const float* __restrict__ Wj,
                      const float* __restrict__ bj, int N) {
  int i = blockIdx.x * blockDim.x + threadIdx.x;
  if (i >= N) return;
  float acc[NCLS];
#pragma unroll
  for (int c = 0; c < NCLS; ++c) acc[c] = bj[c];
#pragma unroll 4
  for (int k = 0; k < HID; ++k) {
    float v = x0[(size_t)i * HID + k];
#pragma unroll
    for (int c = 0; c < NCLS; ++c) acc[c] += v * Wj[k * NCLS + c];
  }
#pragma unroll 4
  for (int k = 0; k < HID; ++k) {
    float v = x1[(size_t)i * HID + k];
#pragma unroll
    for (int c = 0; c < NCLS; ++c) acc[c] += v * Wj[(HID + k) * NCLS + c];
  }
  float mx = acc[0];
#pragma unroll
  for (int c = 1; c < NCLS; ++c) mx = fmaxf(mx, acc[c]);
  float se = 0.0f;
#pragma unroll
  for (int c = 0; c < NCLS; ++c) se += expf(acc[c] - mx);
  float lse = mx + logf(se);
#pragma unroll
  for (int c = 0; c < NCLS; ++c) out[(size_t)i * NCLS + c] = acc[c] - lse;
}

// ---------------------------------------------------------------------------
extern "C" void kernel_launch(void* const* d_in, const int* in_sizes, int n_in,
                              void* d_out, int out_size, void* d_ws, size_t ws_size,
                              hipStream_t stream) {
  const float* b   = (const float*)d_in[0];
  const float* x   = (const float*)d_in[1];
  const int*   ei  = (const int*)d_in[2];
  const float* W0  = (const float*)d_in[3];
  const float* bc0 = (const float*)d_in[4];
  const float* W1  = (const float*)d_in[5];
  const float* bc1 = (const float*)d_in[6];
  const float* g0  = (const float*)d_in[7];
  const float* be0 = (const float*)d_in[8];
  const float* g1  = (const float*)d_in[9];
  const float* be1 = (const float*)d_in[10];
  const float* Wa0 = (const float*)d_in[11];
  const float* ba0 = (const float*)d_in[12];
  const float* Wa1 = (const float*)d_in[13];
  const float* ba1 = (const float*)d_in[14];
  const float* Wa2 = (const float*)d_in[15];
  const float* ba2 = (const float*)d_in[16];
  const float* Wa3 = (const float*)d_in[17];
  const float* ba3 = (const float*)d_in[18];
  const float* Wj  = (const float*)d_in[19];
  const float* bj  = (const float*)d_in[20];

  const int N = N_NODES, H = HID;
  const int E = in_sizes[2] / 2;
  const int* src = ei;
  const int* dst = ei + E;

  float* w    = (float*)d_ws;
  float* dinv = w;  w += N;
  float* htmp = w;  w += (size_t)N * H;
  float* tg   = w;  w += (size_t)N * H;
  float* agg  = w;  w += (size_t)N * H;
  float* x0   = w;  w += (size_t)N * H;
  float* x1   = w;  w += (size_t)N * H;
  float* b0v  = w;  w += (size_t)N * H;
  float* b1v  = w;  w += (size_t)N * H;
  float* b2v  = w;  w += (size_t)N * H;

  float* outLS = (float*)d_out;
  float* outB  = (float*)d_out + (size_t)N * NCLS;

  const int NH = N * H;
  const unsigned EH_blocks = (unsigned)(((size_t)E * H + 255) / 256);

  // degree -> dinv (self loops included)
  k_zero<<<(N + 255) / 256, 256, 0, stream>>>(dinv, N);
  k_deg<<<(E + 255) / 256, 256, 0, stream>>>(dinv, dst, E);
  k_dinv<<<(N + 255) / 256, 256, 0, stream>>>(dinv, N);

  // ---- layer 0: h = x @ W0 ; agg ; BN ; relu -> x0
  k_gemm<128, 2, false><<<dim3(N / 16, 1, 1), 32, 0, stream>>>(
      htmp, x, W0, nullptr, H, F_IN_DIM);
  k_self<<<(NH + 255) / 256, 256, 0, stream>>>(agg, htmp, dinv, bc0, N, H);
  k_edge<<<EH_blocks, 256, 0, stream>>>(agg, htmp, dinv, src, dst, E, H);
  k_bn_relu<<<H, 256, 0, stream>>>(x0, agg, g0, be0, N, H);

  // ---- b0 = b @ Wa0 + ba0  (big GEMM #1: 268MB stream, split-K=32,
  //      16x32 tile per wave so b is read exactly once)
  k_bias_fill<<<(NH + 255) / 256, 256, 0, stream>>>(b0v, ba0, N, H);
  k_gemm<256, 2, true><<<dim3(N / 16, 1, 32), 32, 0, stream>>>(
      b0v, b, Wa0, nullptr, H, N);

  // ---- h = x0 + b0
  k_add<<<(NH + 255) / 256, 256, 0, stream>>>(htmp, x0, b0v, NH);

  // ---- layer 1: t = h @ W1 ; agg ; BN ; relu -> x1
  k_gemm<32, 2, false><<<dim3(N / 16, 1, 1), 32, 0, stream>>>(
      tg, htmp, W1, nullptr, H, H);
  k_self<<<(NH + 255) / 256, 256, 0, stream>>>(agg, tg, dinv, bc1, N, H);
  k_edge<<<EH_blocks, 256, 0, stream>>>(agg, tg, dinv, src, dst, E, H);
  k_bn_relu<<<H, 256, 0, stream>>>(x1, agg, g1, be1, N, H);

  // ---- decoder tail: b1 = b0@Wa1+ba1 ; b2 = b1@Wa2+ba2 ; b3 = b2@Wa3+ba3
  k_gemm<32, 2, false><<<dim3(N / 16, 1, 1), 32, 0, stream>>>(
      b1v, b0v, Wa1, ba1, H, H);
  k_gemm<32, 2, false><<<dim3(N / 16, 1, 1), 32, 0, stream>>>(
      b2v, b1v, Wa2, ba2, H, H);
  // big GEMM #2: [8192,32] x [32,8192] -> d_out tail (store-bound, K unrolled)
  k_gemm<32, 1, false><<<dim3(N / 16, N / 16, 1), 32, 0, stream>>>(
      outB, b2v, Wa3, ba3, N, H);

  // ---- jump classifier + log_softmax -> d_out head
  k_cls<<<(N + 255) / 256, 256, 0, stream>>>(outLS, x0, x1, Wj, bj, N);
}